// neural_spline_flow_37512244363278
// MI455X (gfx1250) — compile-verified
//
#include <hip/hip_runtime.h>
#include <math.h>

typedef __attribute__((ext_vector_type(2))) float v2f;
typedef __attribute__((ext_vector_type(8))) float v8f;

#define BBOUND 3.0f
#define KBINS 10
#define SD 8
#define TDIM 8
#define HID 50
#define TOTALD 16
#define NROWS 262144
#define WAVES_PER_BLOCK 2
#define ROWS_PER_WAVE 16

// per-wave LDS layout (floats)
#define BUFA_OFF 0
#define BUFW_OFF (16 * 64)
#define BUFH_OFF (BUFW_OFF + 16 * 80)
#define BUFD_OFF (BUFH_OFF + 16 * 80)
#define WAVE_SMEM (BUFD_OFF + 16 * 80)   // 4864 floats = 19456 B per wave

__device__ __forceinline__ v8f wmma_step(v8f acc, v2f a, v2f b) {
    // D = A(16x4 f32) * B(4x16 f32) + C(16x16 f32)
    return __builtin_amdgcn_wmma_f32_16x16x4_f32(
        false, a, false, b, (short)0, acc, false, false);
}

// B fragment: element (k,n) -> lane = (n%16) + 16*(k>=2), vgpr = k%2
__device__ __forceinline__ v2f load_b_frag(const float* __restrict__ W,
                                           int Krows, int Ncols,
                                           int kb, int n0, int lane) {
    int n = n0 + (lane & 15);
    int k = kb + ((lane >> 4) << 1);
    v2f b;
    b.x = ((k     < Krows) && (n < Ncols)) ? W[k * Ncols + n]       : 0.0f;
    b.y = ((k + 1 < Krows) && (n < Ncols)) ? W[(k + 1) * Ncols + n] : 0.0f;
    return b;
}

// Hardware reciprocal: single v_rcp_f32 (TRANS32) instead of the ~10-op
// IEEE division sequence (v_div_scale/v_div_fmas/v_div_fixup).
__device__ __forceinline__ float fast_rcp(float v) {
    return __builtin_amdgcn_rcpf(v);
}

// Branchless fast transcendentals: single v_exp_f32 / v_log_f32 / v_rcp_f32
// (TRANS32, co-execute with VALU on CDNA5) instead of branchy libm expansions.
__device__ __forceinline__ float fast_tanh(float x) {
    // tanh(x) = sign(x) * (1 - 2/(1 + e^{2|x|})); exp overflow -> 1 (correct)
    float e = __expf(2.0f * fabsf(x));
    float t = 1.0f - 2.0f * fast_rcp(1.0f + e);
    return copysignf(t, x);
}

__device__ __forceinline__ float fast_softplus(float v) {
    return fmaxf(v, 0.0f) + __logf(1.0f + __expf(-fabsf(v)));
}

__global__ __launch_bounds__(WAVES_PER_BLOCK * 32)
void nsf_kernel(const float* __restrict__ x,
                const float* __restrict__ w0, const float* __restrict__ b0,
                const float* __restrict__ w1, const float* __restrict__ b1,
                const float* __restrict__ ww, const float* __restrict__ bw,
                const float* __restrict__ wh, const float* __restrict__ bh,
                const float* __restrict__ wd, const float* __restrict__ bd,
                float* __restrict__ out, float* __restrict__ partial) {
    __shared__ float smem[WAVES_PER_BLOCK * WAVE_SMEM];
    __shared__ float red[WAVES_PER_BLOCK * 32];

    const int tid  = threadIdx.x;
    const int wave = tid >> 5;
    const int lane = tid & 31;

    float* sw_   = smem + wave * WAVE_SMEM;
    float* bufA  = sw_ + BUFA_OFF;
    float* bufW  = sw_ + BUFW_OFF;
    float* bufH  = sw_ + BUFH_OFF;
    float* bufD  = sw_ + BUFD_OFF;

    const int tile = blockIdx.x * WAVES_PER_BLOCK + wave;
    const int row0 = tile * ROWS_PER_WAVE;

    const int m  = lane & 15;            // A-matrix row (M)
    const int kh = (lane >> 4) << 1;     // A/B k-sublane: 0 or 2
    const int n  = lane & 15;            // B/C column
    const int mb = (lane >> 4) << 3;     // C rows base: 0 or 8

    // ---------- Phase 1: H0 = tanh(zd @ w0 + b0);  zd = x[:, 0:8] ----------
    {
        #pragma unroll
        for (int nt = 0; nt < 4; ++nt) {
            const int n0 = nt * 16;
            const float bv = (n0 + n < HID) ? b0[n0 + n] : 0.0f;
            v8f acc = {bv, bv, bv, bv, bv, bv, bv, bv};
            #pragma unroll
            for (int kb = 0; kb < SD; kb += 4) {
                const float* xr = x + (size_t)(row0 + m) * TOTALD + kb + kh;
                v2f a; a.x = xr[0]; a.y = xr[1];
                v2f b = load_b_frag(w0, SD, HID, kb, n0, lane);
                acc = wmma_step(acc, a, b);
            }
            #pragma unroll
            for (int i = 0; i < 8; ++i) {
                float v = (n0 + n < HID) ? fast_tanh(acc[i]) : 0.0f;
                bufA[(mb + i) * 64 + n0 + n] = v;   // cols >=50 zero-padded
            }
        }
    }
    __syncthreads();

    // ---------- Phase 2: H1 = tanh(H0 @ w1 + b1) ----------
    {
        v8f acc[4];
        #pragma unroll
        for (int nt = 0; nt < 4; ++nt) {
            const int n0 = nt * 16;
            const float bv = (n0 + n < HID) ? b1[n0 + n] : 0.0f;
            acc[nt] = (v8f){bv, bv, bv, bv, bv, bv, bv, bv};
        }
        for (int kb = 0; kb < 52; kb += 4) {   // K padded 50 -> 52 (zeros)
            v2f a = *(const v2f*)&bufA[m * 64 + kb + kh];
            #pragma unroll
            for (int nt = 0; nt < 4; ++nt) {
                v2f b = load_b_frag(w1, HID, HID, kb, nt * 16, lane);
                acc[nt] = wmma_step(acc[nt], a, b);
            }
        }
        #pragma unroll
        for (int nt = 0; nt < 4; ++nt) {
            const int n0 = nt * 16;
            #pragma unroll
            for (int i = 0; i < 8; ++i) {
                float v = (n0 + n < HID) ? fast_tanh(acc[nt][i]) : 0.0f;
                bufA[(mb + i) * 64 + n0 + n] = v;
            }
        }
    }
    __syncthreads();

    // ---------- Phases 3-5: logits heads ----------
    auto gemm_logits = [&](const float* __restrict__ W,
                           const float* __restrict__ bias,
                           int Ncols, float* __restrict__ bufOut, float scale) {
        v8f acc[5];
        #pragma unroll
        for (int nt = 0; nt < 5; ++nt) {
            const int n0 = nt * 16;
            const float bv = (n0 + n < Ncols) ? bias[n0 + n] : 0.0f;
            acc[nt] = (v8f){bv, bv, bv, bv, bv, bv, bv, bv};
        }
        for (int kb = 0; kb < 52; kb += 4) {
            v2f a = *(const v2f*)&bufA[m * 64 + kb + kh];
            #pragma unroll
            for (int nt = 0; nt < 5; ++nt) {
                v2f b = load_b_frag(W, HID, Ncols, kb, nt * 16, lane);
                acc[nt] = wmma_step(acc[nt], a, b);
            }
        }
        #pragma unroll
        for (int nt = 0; nt < 5; ++nt) {
            const int n0 = nt * 16;
            #pragma unroll
            for (int i = 0; i < 8; ++i) {
                float v = (n0 + n < Ncols) ? scale * acc[nt][i] : 0.0f;
                bufOut[(mb + i) * 80 + n0 + n] = v;
            }
        }
    };
    gemm_logits(ww, bw, TDIM * KBINS,       bufW, 2.0f * BBOUND);
    gemm_logits(wh, bh, TDIM * KBINS,       bufH, 2.0f * BBOUND);
    gemm_logits(wd, bd, TDIM * (KBINS - 1), bufD, 1.0f);
    __syncthreads();

    // ---------- Phase 6: per-(row,dim) spline ----------
    float ldsum = 0.0f;
    const float twoB = 2.0f * BBOUND;
    #pragma unroll
    for (int g = 0; g < 4; ++g) {
        const int p = g * 32 + lane;
        const int r = p >> 3;   // 0..15
        const int d = p & 7;    // 0..7

        const float* lw = &bufW[r * 80 + d * KBINS];
        const float* lh = &bufH[r * 80 + d * KBINS];
        const float* lg = &bufD[r * 80 + d * (KBINS - 1)];

        float tw[KBINS], th[KBINS], td[KBINS - 1];
        #pragma unroll
        for (int j = 0; j < KBINS; ++j) { tw[j] = lw[j]; th[j] = lh[j]; }
        #pragma unroll
        for (int j = 0; j < KBINS - 1; ++j) { td[j] = lg[j]; }

        // stable softmax (hardware exp)
        float mw = tw[0], mh = th[0];
        #pragma unroll
        for (int j = 1; j < KBINS; ++j) { mw = fmaxf(mw, tw[j]); mh = fmaxf(mh, th[j]); }
        float sumw = 0.0f, sumh = 0.0f;
        #pragma unroll
        for (int j = 0; j < KBINS; ++j) {
            tw[j] = __expf(tw[j] - mw); sumw += tw[j];
            th[j] = __expf(th[j] - mh); sumh += th[j];
        }
        const float invw = fast_rcp(sumw), invh = fast_rcp(sumh);

        const size_t xrow = (size_t)(row0 + r) * TOTALD;
        const float u = x[xrow + SD + d];
        const bool inside = (u > -BBOUND) && (u < BBOUND);
        const float uc = fminf(fmaxf(u, -BBOUND), BBOUND);

        // fused cumsum + bin select (monotone condition, branchless)
        float xe = -BBOUND, ye = -BBOUND;
        float xk = -BBOUND, yk = -BBOUND, wk = 1.0f, hk = 1.0f;
        float dk = 1.0f, dk1 = 1.0f;
        float dprev = 1.0f;  // d[0] = 1
        #pragma unroll
        for (int j = 0; j < KBINS; ++j) {
            const float wj = twoB * tw[j] * invw;
            const float hj = twoB * th[j] * invh;
            const float dnext = (j < KBINS - 1) ? fast_softplus(td[j]) : 1.0f; // d[j+1]
            const bool sel = (uc >= xe);
            xk  = sel ? xe    : xk;
            yk  = sel ? ye    : yk;
            wk  = sel ? wj    : wk;
            hk  = sel ? hj    : hk;
            dk  = sel ? dprev : dk;
            dk1 = sel ? dnext : dk1;
            xe += wj; ye += hj;
            dprev = dnext;
        }

        const float rwk = fast_rcp(wk);
        const float sk = hk * rwk;
        const float xi = (uc - xk) * rwk;
        const float om = 1.0f - xi;
        const float denom = sk + (dk1 + dk - 2.0f * sk) * xi * om;
        const float rden  = fast_rcp(denom);
        const float num   = dk1 * xi * xi + 2.0f * sk * xi * om + dk * om * om;
        const float y = yk + hk * (sk * xi * xi + dk * xi * om) * rden;
        // 2*log(sk) + log(num) - 2*log(denom) == log(sk^2 * num / denom^2)
        const float logdet = __logf((sk * sk) * num * (rden * rden));

        const float zD  = inside ? y : u;
        const float ldv = inside ? logdet : 0.0f;
        ldsum += ldv;

        out[xrow + d]      = x[xrow + d];  // passthrough zd
        out[xrow + SD + d] = zD;
    }

    // deterministic per-block partial sum
    red[tid] = ldsum;
    __syncthreads();
    if (tid == 0) {
        float s = 0.0f;
        for (int i = 0; i < WAVES_PER_BLOCK * 32; ++i) s += red[i];
        partial[blockIdx.x] = s;
    }
}

__global__ __launch_bounds__(256)
void nsf_reduce_kernel(const float* __restrict__ partial, int nblocks,
                       float* __restrict__ out_scalar) {
    __shared__ float red[256];
    const int tid = threadIdx.x;
    float s = 0.0f;
    for (int i = tid; i < nblocks; i += 256) s += partial[i];
    red[tid] = s;
    __syncthreads();
    for (int off = 128; off > 0; off >>= 1) {
        if (tid < off) red[tid] += red[tid + off];
        __syncthreads();
    }
    if (tid == 0) *out_scalar = red[0];
}

extern "C" void kernel_launch(void* const* d_in, const int* in_sizes, int n_in,
                              void* d_out, int out_size, void* d_ws, size_t ws_size,
                              hipStream_t stream) {
    const float* x  = (const float*)d_in[0];
    const float* w0 = (const float*)d_in[1];
    const float* b0 = (const float*)d_in[2];
    const float* w1 = (const float*)d_in[3];
    const float* b1 = (const float*)d_in[4];
    const float* ww = (const float*)d_in[5];
    const float* bw = (const float*)d_in[6];
    const float* wh = (const float*)d_in[7];
    const float* bh = (const float*)d_in[8];
    const float* wd = (const float*)d_in[9];
    const float* bd = (const float*)d_in[10];
    float* out = (float*)d_out;
    float* partial = (float*)d_ws;  // 8192 floats = 32 KB scratch

    const int tiles  = NROWS / ROWS_PER_WAVE;       // 16384
    const int blocks = tiles / WAVES_PER_BLOCK;     // 8192

    nsf_kernel<<<blocks, WAVES_PER_BLOCK * 32, 0, stream>>>(
        x, w0, b0, w1, b1, ww, bw, wh, bh, wd, bd, out, partial);
    nsf_reduce_kernel<<<1, 256, 0, stream>>>(
        partial, blocks, out + (size_t)NROWS * TOTALD);
}